// GATModel_28037546508297
// MI455X (gfx1250) — compile-verified
//
#include <hip/hip_runtime.h>
#include <hip/hip_bf16.h>

typedef float v2f __attribute__((ext_vector_type(2)));
typedef float v8f __attribute__((ext_vector_type(8)));

#define HC 256   // H*C
#define CH 64    // C
#define NH 4     // heads

// ---------------------------------------------------------------------------
// Dense GEMM  Hout[N,256] = X[N,K] @ W[K,256]  using V_WMMA_F32_16X16X4_F32.
// One wave computes one 16x16 output tile; K stepped by 4 per WMMA.
// fp32 in / fp32 accumulate -> bit-compatible with the fp32 reference GEMM.
// ---------------------------------------------------------------------------
__global__ void gat_gemm_wmma_f32(const float* __restrict__ X,
                                  const float* __restrict__ W,
                                  float* __restrict__ Hout,
                                  int Nrows, int K) {
  const int wave = (blockIdx.x * blockDim.x + threadIdx.x) >> 5;
  const int lane = threadIdx.x & 31;
  const int rowTile = wave >> 4;        // 16 col tiles (256/16)
  const int colTile = wave & 15;
  if (rowTile * 16 >= Nrows) return;    // wave-uniform guard (EXEC stays all-1s)

  const int row0 = rowTile * 16;
  const int col0 = colTile * 16;
  const int hi   = lane >> 4;           // 0 = lanes 0-15, 1 = lanes 16-31
  const int l15  = lane & 15;

  // A fragment: lane holds row (row0+l15); VGPR0=K+ (hi?2:0), VGPR1=+1
  int arow = row0 + l15;
  if (arow >= Nrows) arow = Nrows - 1;  // clamp (exact tiling for N=50000 anyway)
  const int koffA = hi << 1;
  // B fragment: VGPR0 -> K = k0 + (hi?2:0), VGPR1 -> +1 ; N = col0 + l15
  const int bcol = col0 + l15;

  v8f c = {};
  for (int k0 = 0; k0 < K; k0 += 4) {
    v2f a, b;
    const float* ap = X + (size_t)arow * K + k0 + koffA;
    a.x = ap[0];
    a.y = ap[1];
    const float* bp = W + (size_t)(k0 + koffA) * HC + bcol;
    b.x = bp[0];
    b.y = bp[HC];
    c = __builtin_amdgcn_wmma_f32_16x16x4_f32(false, a, false, b,
                                              (short)0, c, false, false);
  }

  // C/D layout: VGPR r -> row (row0 + r + hi*8), col (col0 + l15)
  const int srow = row0 + (hi << 3);
#pragma unroll
  for (int r = 0; r < 8; ++r) {
    if (srow + r < Nrows)
      Hout[(size_t)(srow + r) * HC + col0 + l15] = c[r];
  }
}

// ---------------------------------------------------------------------------
// Per-node attention logits: asrc[n,h] = sum_c h[n,h,c]*a_src[h,c] (same dst)
// ---------------------------------------------------------------------------
__global__ void gat_node_alpha(const float* __restrict__ Hbuf,
                               const float* __restrict__ a_src,
                               const float* __restrict__ a_dst,
                               float* __restrict__ asrc,
                               float* __restrict__ adst, int n) {
  int t = blockIdx.x * blockDim.x + threadIdx.x;
  if (t >= n * NH) return;
  const int node = t >> 2, h = t & 3;
  const float* hp = Hbuf + (size_t)node * HC + h * CH;
  const float* as = a_src + h * CH;
  const float* ad = a_dst + h * CH;
  float s0 = 0.f, s1 = 0.f;
#pragma unroll 8
  for (int c = 0; c < CH; ++c) {
    const float v = hp[c];
    s0 += v * as[c];
    s1 += v * ad[c];
  }
  asrc[t] = s0;
  adst[t] = s1;
}

__global__ void gat_fill(float* __restrict__ p, float v, long long n) {
  long long i = (long long)blockIdx.x * blockDim.x + threadIdx.x;
  if (i < n) p[i] = v;
}

__device__ __forceinline__ float leaky(float v) { return v > 0.f ? v : 0.2f * v; }

__device__ __forceinline__ void atomicMaxFloat(float* addr, float val) {
  if (val >= 0.f)
    atomicMax((int*)addr, __float_as_int(val));
  else
    atomicMin((unsigned int*)addr, (unsigned int)__float_as_int(val));
}

// ---------------------------------------------------------------------------
// Edge pass 1: segment max of leakyrelu(asrc[src]+adst[dst]) over dst
// ---------------------------------------------------------------------------
__global__ void gat_edge_max(const int* __restrict__ ei, int E, int Etot,
                             const float* __restrict__ asrc,
                             const float* __restrict__ adst,
                             float* __restrict__ emax) {
  int t = blockIdx.x * blockDim.x + threadIdx.x;
  if (t >= Etot * NH) return;
  const int e = t >> 2, h = t & 3;
  int s, d;
  if (e < E) { s = ei[e]; d = ei[E + e]; }
  else       { s = d = e - E; }          // self loops
  const float v = leaky(asrc[s * NH + h] + adst[d * NH + h]);
  atomicMaxFloat(&emax[d * NH + h], v);
}

// ---------------------------------------------------------------------------
// Edge pass 2 (fused): ex = exp(e - max); denom[dst] += ex;
// accum[dst,h,c] += h[src,h,c] * ex.  One wave per edge; lane covers 8 of the
// 256 message floats.  Division by denom is hoisted to the finalize kernel.
// ---------------------------------------------------------------------------
__global__ void gat_edge_accum(const int* __restrict__ ei, int E, int Etot,
                               const float* __restrict__ asrc,
                               const float* __restrict__ adst,
                               const float* __restrict__ emax,
                               const float* __restrict__ Hbuf,
                               float* __restrict__ denom,
                               float* __restrict__ accum) {
  const int wave = (blockIdx.x * blockDim.x + threadIdx.x) >> 5;
  const int lane = threadIdx.x & 31;
  if (wave >= Etot) return;
  int s, d;
  if (wave < E) { s = ei[wave]; d = ei[E + wave]; }
  else          { s = d = wave - E; }

  float ex[NH];
#pragma unroll
  for (int h = 0; h < NH; ++h) {
    const float v = leaky(asrc[s * NH + h] + adst[d * NH + h]);
    ex[h] = __expf(v - emax[d * NH + h]);
  }
  if (lane == 0) {
#pragma unroll
    for (int h = 0; h < NH; ++h) atomicAdd(&denom[d * NH + h], ex[h]);
  }
  const float* hs = Hbuf + (size_t)s * HC;
  float* ac = accum + (size_t)d * HC;
#pragma unroll
  for (int i = 0; i < 8; ++i) {
    const int p = lane + (i << 5);
    atomicAdd(&ac[p], hs[p] * ex[p >> 6]);
  }
}

// ---------------------------------------------------------------------------
// Finalize: x_out[n,c] = relu( mean_h(accum[n,h,c]/(denom[n,h]+1e-16)) + b[c])
// ---------------------------------------------------------------------------
__global__ void gat_finalize(const float* __restrict__ accum,
                             const float* __restrict__ denom,
                             const float* __restrict__ bias,
                             float* __restrict__ Xout, int n) {
  int t = blockIdx.x * blockDim.x + threadIdx.x;
  if (t >= n * CH) return;
  const int node = t >> 6, c = t & 63;
  float acc = 0.f;
#pragma unroll
  for (int h = 0; h < NH; ++h)
    acc += accum[(size_t)node * HC + h * CH + c] / (denom[node * NH + h] + 1e-16f);
  const float v = acc * 0.25f + bias[c];
  Xout[t] = v > 0.f ? v : 0.f;
}

// ---------------------------------------------------------------------------
// Global mean pool + linear head
// ---------------------------------------------------------------------------
__global__ void gat_pool(const float* __restrict__ X, const int* __restrict__ batch,
                         float* __restrict__ pool, float* __restrict__ cnt, int n) {
  int t = blockIdx.x * blockDim.x + threadIdx.x;
  if (t >= n * CH) return;
  const int node = t >> 6, c = t & 63;
  const int g = batch[node];
  atomicAdd(&pool[(size_t)g * CH + c], X[t]);
  if (c == 0) atomicAdd(&cnt[g], 1.0f);
}

__global__ void gat_head(const float* __restrict__ pool, const float* __restrict__ cnt,
                         const float* __restrict__ lw, const float* __restrict__ lb,
                         float* __restrict__ out, int G) {
  int g = blockIdx.x * blockDim.x + threadIdx.x;
  if (g >= G) return;
  const float inv = 1.0f / fmaxf(cnt[g], 1.0f);
  float acc = lb[0];
  const float* pp = pool + (size_t)g * CH;
#pragma unroll 8
  for (int c = 0; c < CH; ++c) acc += pp[c] * inv * lw[c];
  out[g] = acc;
}

// ---------------------------------------------------------------------------
extern "C" void kernel_launch(void* const* d_in, const int* in_sizes, int n_in,
                              void* d_out, int out_size, void* d_ws, size_t ws_size,
                              hipStream_t stream) {
  const float* x0    = (const float*)d_in[0];
  const int*   ei    = (const int*)d_in[1];
  const int*   batch = (const int*)d_in[2];
  const float* W[3]    = {(const float*)d_in[3],  (const float*)d_in[7],  (const float*)d_in[11]};
  const float* Asrc[3] = {(const float*)d_in[4],  (const float*)d_in[8],  (const float*)d_in[12]};
  const float* Adst[3] = {(const float*)d_in[5],  (const float*)d_in[9],  (const float*)d_in[13]};
  const float* Bias[3] = {(const float*)d_in[6],  (const float*)d_in[10], (const float*)d_in[14]};
  const float* lin_w = (const float*)d_in[15];
  const float* lin_b = (const float*)d_in[16];
  float* out = (float*)d_out;

  const int F_IN = 128;
  const int N    = in_sizes[0] / F_IN;
  const int E    = in_sizes[1] / 2;
  const int Etot = E + N;
  const int G    = out_size;

  // workspace layout (floats)
  float* ws = (float*)d_ws;
  size_t off = 0;
  float* hbuf  = ws + off; off += (size_t)N * HC;
  float* accum = ws + off; off += (size_t)N * HC;
  float* asrc  = ws + off; off += (size_t)N * NH;
  float* adst  = ws + off; off += (size_t)N * NH;
  float* emax  = ws + off; off += (size_t)N * NH;
  float* denom = ws + off; off += (size_t)N * NH;
  float* xA    = ws + off; off += (size_t)N * CH;
  float* xB    = ws + off; off += (size_t)N * CH;
  float* pool  = ws + off; off += (size_t)G * CH;
  float* cnt   = ws + off; off += (size_t)G;
  (void)ws_size; (void)n_in;

  const dim3 blk(256);
  auto nblk = [](long long threads) { return dim3((unsigned)((threads + 255) / 256)); };

  const float* xin = x0;
  float* xout = xA;
  for (int L = 0; L < 3; ++L) {
    const int K = (L == 0) ? F_IN : CH;
    const long long rowTiles = (N + 15) / 16;
    const long long gemmThreads = rowTiles * 16 /*col tiles*/ * 32;
    gat_gemm_wmma_f32<<<nblk(gemmThreads), blk, 0, stream>>>(xin, W[L], hbuf, N, K);

    gat_node_alpha<<<nblk((long long)N * NH), blk, 0, stream>>>(hbuf, Asrc[L], Adst[L],
                                                                asrc, adst, N);
    gat_fill<<<nblk((long long)N * NH), blk, 0, stream>>>(emax, -3.0e38f, (long long)N * NH);
    gat_fill<<<nblk((long long)N * NH), blk, 0, stream>>>(denom, 0.f, (long long)N * NH);
    gat_fill<<<nblk((long long)N * HC), blk, 0, stream>>>(accum, 0.f, (long long)N * HC);

    gat_edge_max<<<nblk((long long)Etot * NH), blk, 0, stream>>>(ei, E, Etot, asrc, adst, emax);
    gat_edge_accum<<<nblk((long long)Etot * 32), blk, 0, stream>>>(ei, E, Etot, asrc, adst,
                                                                   emax, hbuf, denom, accum);
    gat_finalize<<<nblk((long long)N * CH), blk, 0, stream>>>(accum, denom, Bias[L], xout, N);

    xin  = xout;
    xout = (xout == xA) ? xB : xA;
  }

  gat_fill<<<nblk((long long)G * CH), blk, 0, stream>>>(pool, 0.f, (long long)G * CH);
  gat_fill<<<nblk((long long)G), blk, 0, stream>>>(cnt, 0.f, (long long)G);
  gat_pool<<<nblk((long long)N * CH), blk, 0, stream>>>(xin, batch, pool, cnt, N);
  gat_head<<<nblk((long long)G), blk, 0, stream>>>(pool, cnt, lin_w, lin_b, out, G);
}